// GatedBlockParity_63187558859316
// MI455X (gfx1250) — compile-verified
//
#include <hip/hip_runtime.h>
#include <cstdint>

typedef float v4f __attribute__((ext_vector_type(4)));

namespace {
constexpr int kDS     = 256;            // scalar block
constexpr int kDG     = 384;            // gate block
constexpr int kInW    = 2048;           // 256 + 384 + 768 + 640
constexpr int kOutW   = 1664;           // 256 + 768 + 640
constexpr int kScal4  = kDS / 4;        // 64  float4 of scalars
constexpr int kGate4  = kDG / 4;        // 96  float4 of gates
constexpr int kNs14   = 768 / 4;        // 192 float4 of dim-3 nonscalars
constexpr int kNs4    = (768 + 640)/4;  // 352 float4 of nonscalars total
constexpr int kIn4Ns  = (kDS + kDG)/4;  // 160: nonscalar start (input, float4)
constexpr int kOut4Ns = kDS / 4;        // 64 : nonscalar start (output, float4)
constexpr int kThreads = kNs4;          // 352 threads = 11 wave32, fully uniform
}

__device__ __forceinline__ float sigmoid1(float x) {
  // sigmoid(x) = 1 / (1 + exp(-x)); exp(-x) = exp2(-x * log2(e))
  float e = __builtin_amdgcn_exp2f(x * -1.4426950408889634f);
  return __builtin_amdgcn_rcpf(1.0f + e);
}

extern "C" __global__ __launch_bounds__(kThreads)
void gated_block_parity_kernel(const float* __restrict__ in,
                               float* __restrict__ out) {
  __shared__ __align__(16) float sg[kDG];   // sigmoided gates for this row

  const int t = threadIdx.x;
  const float* __restrict__ in_row  = in  + (size_t)blockIdx.x * kInW;
  float* __restrict__       out_row = out + (size_t)blockIdx.x * kOutW;
  const v4f* __restrict__ in4  = (const v4f*)in_row;
  v4f* __restrict__       out4 = (v4f*)out_row;

  // ---- Stage 0: one unconditional NT b128 nonscalar load per thread.
  // Issued before the gate work so HBM latency hides behind the async DMA,
  // sigmoid, and barrier. No conditionals -> no phi -> no early waitcnt.
  v4f ns = __builtin_nontemporal_load(in4 + kIn4Ns + t);

  if (t < kGate4) {
    // ---- waves 0..2: async DMA raw gates (1536 B) global -> LDS, b128/lane
    unsigned lds_addr = (unsigned)(uintptr_t)(&sg[0]) + (unsigned)(t * 16);
    const float* gsrc = in_row + kDS + t * 4;
    asm volatile(
        "global_load_async_to_lds_b128 %0, %1, off\n\t"
        "s_wait_asynccnt 0x0"
        :
        : "v"(lds_addr), "v"(gsrc)
        : "memory");
    // sigmoid in place in LDS
    v4f g = ((v4f*)sg)[t];
    g.x = sigmoid1(g.x); g.y = sigmoid1(g.y);
    g.z = sigmoid1(g.z); g.w = sigmoid1(g.w);
    ((v4f*)sg)[t] = g;
  } else if (t < kGate4 + kScal4) {
    // ---- waves 3..4 (64 lanes): ReLU scalar block, streaming NT b128
    const int j = t - kGate4;
    v4f v = __builtin_nontemporal_load(in4 + j);
    v.x = fmaxf(v.x, 0.f); v.y = fmaxf(v.y, 0.f);
    v.z = fmaxf(v.z, 0.f); v.w = fmaxf(v.w, 0.f);
    __builtin_nontemporal_store(v, out4 + j);
  }
  __syncthreads();

  // ---- all 11 waves: gate the prefetched nonscalar float4 and store NT.
  // j<192 boundary (t==192) is wave-aligned: no intra-wave divergence.
  if (t < kNs14) {                              // dim-3 groups, gates 0..255
    const unsigned e = 4u * (unsigned)t;
    ns.x *= sg[(e + 0u) / 3u];
    ns.y *= sg[(e + 1u) / 3u];
    ns.z *= sg[(e + 2u) / 3u];
    ns.w *= sg[(e + 3u) / 3u];
  } else {                                      // dim-5 groups, gates 256..383
    const unsigned e = 4u * (unsigned)(t - kNs14);
    ns.x *= sg[256u + (e + 0u) / 5u];
    ns.y *= sg[256u + (e + 1u) / 5u];
    ns.z *= sg[256u + (e + 2u) / 5u];
    ns.w *= sg[256u + (e + 3u) / 5u];
  }
  __builtin_nontemporal_store(ns, out4 + kOut4Ns + t);
}

extern "C" void kernel_launch(void* const* d_in, const int* in_sizes, int n_in,
                              void* d_out, int out_size, void* d_ws, size_t ws_size,
                              hipStream_t stream) {
  (void)n_in; (void)out_size; (void)d_ws; (void)ws_size;
  const float* features = (const float*)d_in[0];
  float* out = (float*)d_out;
  const int rows = in_sizes[0] / kInW;          // 65536
  gated_block_parity_kernel<<<dim3(rows), dim3(kThreads), 0, stream>>>(features, out);
}